// BILSTM_CRF_24824910971199
// MI455X (gfx1250) — compile-verified
//
#include <hip/hip_runtime.h>
#include <math.h>

typedef __attribute__((ext_vector_type(2))) float v2f;
typedef __attribute__((ext_vector_type(8))) float v8f;

// Problem dims: V=100000, E=128, H=256, T=18, B=64, S=512; 4H=1024, K=E+H=384

// ---- workspace layout (in floats) ----
static constexpr size_t OFF_XS = 0;                        // xs[s][b][e]           512*64*128
static constexpr size_t SZ_XS  = 512ull * 64 * 128;
static constexpr size_t OFF_WT = OFF_XS + SZ_XS;           // Wt[d][k][j]           2*384*1024 (transposed combined weights)
static constexpr size_t SZ_WT  = 2ull * 384 * 1024;
static constexpr size_t OFF_BC = OFF_WT + SZ_WT;           // bcomb[d][j]           2*1024
static constexpr size_t SZ_BC  = 2048;
static constexpr size_t OFF_H  = OFF_BC + SZ_BC;           // h state [d][b][h]     2*64*256
static constexpr size_t SZ_H   = 2ull * 64 * 256;
static constexpr size_t OFF_C  = OFF_H + SZ_H;             // c state
static constexpr size_t OFF_G  = OFF_C + SZ_H;             // gates [d][b][j]       2*64*1024
static constexpr size_t SZ_G   = 2ull * 64 * 1024;
static constexpr size_t OFF_FE = OFF_G + SZ_G;             // feats[t][b][tag]      512*64*18
static constexpr size_t SZ_FE  = 512ull * 64 * 18;
static constexpr size_t OFF_BP = OFF_FE + SZ_FE;           // backptr[t][b][tag]    512*64*18 (int32)

// ---------------- prep: gather embeddings, build combined transposed weights ----------------
__global__ void prep_kernel(const int* __restrict__ x,
                            const float* __restrict__ embed,
                            const float* __restrict__ Wih_f, const float* __restrict__ Whh_f,
                            const float* __restrict__ bih_f, const float* __restrict__ bhh_f,
                            const float* __restrict__ Wih_b, const float* __restrict__ Whh_b,
                            const float* __restrict__ bih_b, const float* __restrict__ bhh_b,
                            const float* __restrict__ bc,
                            const float* __restrict__ h0, const float* __restrict__ c0,
                            float* __restrict__ ws) {
  size_t tid = (size_t)blockIdx.x * blockDim.x + threadIdx.x;
  size_t np  = (size_t)gridDim.x * blockDim.x;
  float* xs  = ws + OFF_XS;
  float* Wt  = ws + OFF_WT;
  float* bcm = ws + OFF_BC;
  float* hb  = ws + OFF_H;
  float* cb  = ws + OFF_C;
  float* fe  = ws + OFF_FE;

  for (size_t i = tid; i < SZ_XS; i += np) {          // xs[s][b][e] = embed[x[b][s]][e]
    int e = (int)(i & 127);
    size_t sb = i >> 7;
    int b = (int)(sb & 63);
    int s = (int)(sb >> 6);
    xs[i] = embed[(size_t)x[b * 512 + s] * 128 + e];
  }
  for (size_t i = tid; i < SZ_WT; i += np) {          // Wt[d][k][j] = (k<128 ? Wih : Whh)[j][k]
    int d = (i >= (size_t)384 * 1024) ? 1 : 0;
    size_t r = i - (size_t)d * 384 * 1024;
    int k = (int)(r >> 10);
    int j = (int)(r & 1023);
    const float* Wi = d ? Wih_b : Wih_f;
    const float* Wh = d ? Whh_b : Whh_f;
    Wt[i] = (k < 128) ? Wi[j * 128 + k] : Wh[j * 256 + (k - 128)];
  }
  for (size_t i = tid; i < SZ_BC; i += np) {
    int d = (i >= 1024) ? 1 : 0;
    int j = (int)(i & 1023);
    bcm[i] = d ? (bih_b[j] + bhh_b[j]) : (bih_f[j] + bhh_f[j]);
  }
  for (size_t i = tid; i < SZ_H; i += np) { hb[i] = h0[i]; cb[i] = c0[i]; }
  for (size_t i = tid; i < SZ_FE; i += np) { fe[i] = bc[i % 18]; }
}

// ---------------- per-step GEMM: gates = [x_t | h_prev] @ Wt + bias ----------------
// Each wave computes a full M-strip: 4 tiles of 16x16 (M=64, N=16), sharing one
// B fragment across 4 back-to-back WMMAs. 2 dirs x 64 N-tiles = 128 waves = 16 blocks.
__global__ void lstm_gemm_kernel(int s, float* __restrict__ ws) {
  const float* xs  = ws + OFF_XS;
  const float* Wt  = ws + OFF_WT;
  const float* bcm = ws + OFF_BC;
  const float* hb  = ws + OFF_H;
  float* gates     = ws + OFF_G;

  int lane = threadIdx.x & 31;
  int wave = threadIdx.x >> 5;
  int d    = blockIdx.x >> 3;                    // direction (0..1)
  int nt   = ((blockIdx.x & 7) << 3) + wave;     // N-tile (0..63)
  int t    = d ? (511 - s) : s;

  int mlo = lane & 15;                           // A row within 16-row tile / C column select
  int j   = (nt << 4) + mlo;                     // output column
  int khi = (lane >> 4) << 1;                    // lanes 0-15 -> K 0/1, lanes 16-31 -> K 2/3

  const float* Ax = xs + ((size_t)t * 64 + mlo) * 128 + khi;   // embedded input rows (mt stride 16*128)
  const float* Ah = hb + ((size_t)d * 64 + mlo) * 256 + khi;   // previous hidden rows (mt stride 16*256)
  const float* Wd = Wt + (size_t)d * 384 * 1024 + j;

  v8f acc[4];
  #pragma unroll
  for (int mt = 0; mt < 4; mt++) acc[mt] = v8f{0.f, 0.f, 0.f, 0.f, 0.f, 0.f, 0.f, 0.f};

  #pragma unroll 2
  for (int k0 = 0; k0 < 128; k0 += 4) {          // K = 0..127 : input part
    v2f bm;
    bm.x = Wd[(size_t)(k0 + khi) * 1024];
    bm.y = Wd[(size_t)(k0 + khi + 1) * 1024];
    #pragma unroll
    for (int mt = 0; mt < 4; mt++) {
      v2f a;
      a.x = Ax[mt * 16 * 128 + k0];
      a.y = Ax[mt * 16 * 128 + k0 + 1];
      acc[mt] = __builtin_amdgcn_wmma_f32_16x16x4_f32(false, a, false, bm, (short)0, acc[mt], false, false);
    }
  }
  #pragma unroll 2
  for (int k0 = 0; k0 < 256; k0 += 4) {          // K = 128..383 : recurrent part
    v2f bm;
    bm.x = Wd[(size_t)(128 + k0 + khi) * 1024];
    bm.y = Wd[(size_t)(128 + k0 + khi + 1) * 1024];
    #pragma unroll
    for (int mt = 0; mt < 4; mt++) {
      v2f a;
      a.x = Ah[mt * 16 * 256 + k0];
      a.y = Ah[mt * 16 * 256 + k0 + 1];
      acc[mt] = __builtin_amdgcn_wmma_f32_16x16x4_f32(false, a, false, bm, (short)0, acc[mt], false, false);
    }
  }

  float bias = bcm[d * 1024 + j];
  #pragma unroll
  for (int mt = 0; mt < 4; mt++) {
    int mbase = (mt << 4) + ((lane >> 4) << 3);  // C layout: VGPR v -> M=v (lo lanes) / v+8 (hi lanes)
    #pragma unroll
    for (int v = 0; v < 8; v++) {
      gates[((size_t)d * 64 + mbase + v) * 1024 + j] = acc[mt][v] + bias;
    }
  }
}

// ---------------- per-step pointwise: LSTM cell update + classifier accumulation ----------------
__global__ void lstm_update_kernel(int s, float* __restrict__ ws, const float* __restrict__ Wc) {
  int d  = blockIdx.x >> 6;
  int b  = blockIdx.x & 63;
  int hx = threadIdx.x;                       // 0..255

  float* gates = ws + OFF_G;
  float* hb    = ws + OFF_H;
  float* cb    = ws + OFF_C;
  float* fe    = ws + OFF_FE;

  size_t gbase = ((size_t)d * 64 + b) * 1024;
  float gi = gates[gbase + hx];
  float gf = gates[gbase + 256 + hx];
  float gg = gates[gbase + 512 + hx];
  float go = gates[gbase + 768 + hx];

  size_t hcb = ((size_t)d * 64 + b) * 256 + hx;
  float c  = cb[hcb];
  float si = 1.f / (1.f + expf(-gi));
  float sf = 1.f / (1.f + expf(-gf));
  float so = 1.f / (1.f + expf(-go));
  float cn = sf * c + si * tanhf(gg);
  float hn = so * tanhf(cn);
  cb[hcb] = cn;
  hb[hcb] = hn;

  __shared__ float hsm[256];
  hsm[hx] = hn;
  __syncthreads();

  if (hx < 18) {                              // feats[t][b][tag] += Wc[tag][d*256:+256] . h
    int t = d ? (511 - s) : s;
    const float* wc = Wc + hx * 512 + d * 256;
    float acc = 0.f;
    #pragma unroll 8
    for (int k = 0; k < 256; k++) acc += wc[k] * hsm[k];
    fe[((size_t)t * 64 + b) * 18 + hx] += acc;
  }
}

// ---------------- Viterbi decode (one thread per batch element) ----------------
__global__ void viterbi_kernel(const int* __restrict__ mask,
                               const float* __restrict__ trans,
                               float* __restrict__ ws,
                               float* __restrict__ out) {
  int b = threadIdx.x;
  if (b >= 64) return;
  const float* fe = ws + OFF_FE;
  int* bp = (int*)(ws + OFF_BP);

  int len = 0;
  for (int t = 0; t < 512; t++) len += mask[b * 512 + t];
  int lastPos = len - 1;

  float part[18], lastPart[18];
  for (int k = 0; k < 18; k++) part[k] = fe[(size_t)b * 18 + k] + trans[16 * 18 + k];
  if (lastPos == 0)
    for (int k = 0; k < 18; k++) lastPart[k] = part[k];

  for (int t = 1; t < 512; t++) {
    int m = mask[b * 512 + t];
    const float* ft = fe + ((size_t)t * 64 + b) * 18;
    float newp[18];
    for (int k = 0; k < 18; k++) {
      float best = -3.0e38f;
      int bj = 0;
      for (int jj = 0; jj < 18; jj++) {
        float v = part[jj] + trans[jj * 18 + k];
        if (v > best) { best = v; bj = jj; }   // strict > => first argmax, matches jnp
      }
      newp[k] = best + ft[k];
      bp[((size_t)t * 64 + b) * 18 + k] = (m > 0) ? bj : 0;
    }
    for (int k = 0; k < 18; k++) part[k] = newp[k];
    if (t == lastPos)
      for (int k = 0; k < 18; k++) lastPart[k] = part[k];
  }

  float best = -3.0e38f;
  int pointer = 0;
  for (int jj = 0; jj < 18; jj++) {
    float v = lastPart[jj] + trans[jj * 18 + 17];   // END column
    if (v > best) { best = v; pointer = jj; }
  }
  out[b] = best;                                     // path_score

  float* dec = out + 64 + (size_t)b * 512;
  dec[511] = (float)pointer;
  int ptr = pointer;
  for (int i = 510; i >= 0; i--) {
    int nv;
    if (i > lastPos)       nv = 0;                   // masked bps are zero
    else if (i == lastPos) nv = pointer;             // overwritten backpointer row
    else                   nv = bp[((size_t)(i + 1) * 64 + b) * 18 + ptr];
    dec[i] = (float)nv;
    ptr = nv;
  }
}

extern "C" void kernel_launch(void* const* d_in, const int* in_sizes, int n_in,
                              void* d_out, int out_size, void* d_ws, size_t ws_size,
                              hipStream_t stream) {
  const int*   x     = (const int*)d_in[0];
  const int*   mask  = (const int*)d_in[1];
  const float* embed = (const float*)d_in[2];
  const float* Wih_f = (const float*)d_in[3];
  const float* Whh_f = (const float*)d_in[4];
  const float* bih_f = (const float*)d_in[5];
  const float* bhh_f = (const float*)d_in[6];
  const float* Wih_b = (const float*)d_in[7];
  const float* Whh_b = (const float*)d_in[8];
  const float* bih_b = (const float*)d_in[9];
  const float* bhh_b = (const float*)d_in[10];
  const float* Wc    = (const float*)d_in[11];
  const float* bc    = (const float*)d_in[12];
  const float* trans = (const float*)d_in[13];
  const float* h0    = (const float*)d_in[14];
  const float* c0    = (const float*)d_in[15];
  float* ws  = (float*)d_ws;
  float* out = (float*)d_out;

  prep_kernel<<<1024, 256, 0, stream>>>(x, embed, Wih_f, Whh_f, bih_f, bhh_f,
                                        Wih_b, Whh_b, bih_b, bhh_b, bc, h0, c0, ws);
  for (int s = 0; s < 512; s++) {
    lstm_gemm_kernel<<<16, 256, 0, stream>>>(s, ws);
    lstm_update_kernel<<<128, 256, 0, stream>>>(s, ws, Wc);
  }
  viterbi_kernel<<<1, 64, 0, stream>>>(mask, trans, ws, out);
}